// InteractionAccelerationModule_12481174962511
// MI455X (gfx1250) — compile-verified
//
#include <hip/hip_runtime.h>
#include <math.h>

// -------------------------------------------------------------------------
// obj = sum_{m,k} | (p[m,k]-p[m,k+1]) - (w[m,k]-2w[m,k+1]+w[m,k+2]) | / dt^2
// Memory-bound: 268 MB single pass -> ~11.5 us floor at 23.3 TB/s.
// -------------------------------------------------------------------------

typedef __attribute__((ext_vector_type(2))) float v2f;
typedef __attribute__((ext_vector_type(8))) float v8f;

#define DT2_INV   (1.0f / (0.4f * 0.4f))
#define M_DIM     4096LL
#define T_DIM     4096LL
// Each "pair" j covers outputs k = 2j and 2j+1; valid j in [0, 2047)
#define PAIRS     2047LL
#define NBLOCKS   4096
#define NTHREADS  256

__global__ __launch_bounds__(NTHREADS)
void accnorm_partial_kernel(const float* __restrict__ p,
                            const float* __restrict__ pwo,
                            float* __restrict__ partials)
{
    const long long total  = M_DIM * PAIRS;
    const long long stride = (long long)gridDim.x * blockDim.x;

    float acc = 0.0f;

    for (long long idx = (long long)blockIdx.x * blockDim.x + threadIdx.x;
         idx < total; idx += stride) {
        const long long m    = idx / PAIRS;
        const long long j    = idx - m * PAIRS;
        const long long base = (m * T_DIM + 2 * j) * 2;   // float offset; 16B aligned

        // Coalesced wide loads: b128 + b64 for p, 2x b128 for pwo.
        const float4 p01 = *(const float4*)(p   + base);      // p[k], p[k+1]
        const float2 p2  = *(const float2*)(p   + base + 4);  // p[k+2]
        const float4 w01 = *(const float4*)(pwo + base);      // w[k], w[k+1]
        const float4 w23 = *(const float4*)(pwo + base + 4);  // w[k+2], w[k+3]

        // Speculative prefetch of the next grid-stride iteration's stream
        // (global_prefetch_b8; OOB translations are silently dropped).
        __builtin_prefetch(p   + base + stride * 4, 0, 1);
        __builtin_prefetch(pwo + base + stride * 4, 0, 1);

        // Output k = 2j
        const float dx0 = (p01.x - p01.z) - (w01.x - 2.0f * w01.z + w23.x);
        const float dy0 = (p01.y - p01.w) - (w01.y - 2.0f * w01.w + w23.y);
        // Output k = 2j+1
        const float dx1 = (p01.z - p2.x) - (w01.z - 2.0f * w23.x + w23.z);
        const float dy1 = (p01.w - p2.y) - (w01.w - 2.0f * w23.y + w23.w);

        acc += sqrtf(dx0 * dx0 + dy0 * dy0) + sqrtf(dx1 * dx1 + dy1 * dy1);
    }
    // Norm is homogeneous: divide by dt^2 once per thread instead of per point.
    acc *= DT2_INV;

    // ---- wave32 reduction: 32 lanes -> 2 values via one V_WMMA_F32_16X16X4_F32
    // A (16x4 f32): lane L<16 -> A[L][0]=acc, A[L][1]=0 ; lane L>=16 -> A[L-16][2]=acc.
    // B = all-ones (value 1 in every element => layout-independent).
    // D[m][n] = acc[m] + acc[m+16] for all n. Each lane sums its 8 D VGPRs:
    // lanes 0..15 end with sum(rows 0..7), lanes 16..31 with sum(rows 8..15).
    float s;
#if __has_builtin(__builtin_amdgcn_wmma_f32_16x16x4_f32)
    {
        v2f a; a[0] = acc;  a[1] = 0.0f;
        v2f b; b[0] = 1.0f; b[1] = 1.0f;
        v8f c = {};
        v8f d = __builtin_amdgcn_wmma_f32_16x16x4_f32(
            /*neg_a=*/false, a, /*neg_b=*/false, b,
            /*c_mod=*/(short)0, c, /*reuse_a=*/false, /*reuse_b=*/false);
        s = d[0] + d[1] + d[2] + d[3] + d[4] + d[5] + d[6] + d[7];
    }
#else
    // Fallback: plain shuffle tree (wave32).
    s = acc;
    for (int off = 16; off > 0; off >>= 1)
        s += __shfl_down(s, off, 32);
#endif

    __shared__ float red[(NTHREADS / 32) * 2];
    const unsigned lane = threadIdx.x & 31u;
    const unsigned wave = threadIdx.x >> 5;
#if __has_builtin(__builtin_amdgcn_wmma_f32_16x16x4_f32)
    if (lane == 0u)  red[wave * 2 + 0] = s;   // rows 0..7 half-sum
    if (lane == 16u) red[wave * 2 + 1] = s;   // rows 8..15 half-sum
#else
    if (lane == 0u)  red[wave * 2 + 0] = s;
    if (lane == 16u) red[wave * 2 + 1] = 0.0f;
#endif
    __syncthreads();

    if (threadIdx.x == 0) {
        float t = 0.0f;
#pragma unroll
        for (int i = 0; i < (NTHREADS / 32) * 2; ++i) t += red[i];
        partials[blockIdx.x] = t;
    }
}

__global__ __launch_bounds__(NTHREADS)
void reduce_partials_kernel(const float* __restrict__ partials,
                            float* __restrict__ out)
{
    __shared__ float sm[NTHREADS];
    float a = 0.0f;
    for (int i = threadIdx.x; i < NBLOCKS; i += NTHREADS)
        a += partials[i];
    sm[threadIdx.x] = a;
    __syncthreads();
    for (int off = NTHREADS / 2; off > 0; off >>= 1) {
        if ((int)threadIdx.x < off) sm[threadIdx.x] += sm[threadIdx.x + off];
        __syncthreads();
    }
    if (threadIdx.x == 0) out[0] = sm[0];
}

extern "C" void kernel_launch(void* const* d_in, const int* in_sizes, int n_in,
                              void* d_out, int out_size, void* d_ws, size_t ws_size,
                              hipStream_t stream) {
    (void)in_sizes; (void)n_in; (void)out_size; (void)ws_size;
    const float* p   = (const float*)d_in[0];   // ado_positions    [4096,4096,2] f32
    const float* pwo = (const float*)d_in[1];   // ado_positions_wo [4096,4096,2] f32
    float* out      = (float*)d_out;            // [1] f32
    float* partials = (float*)d_ws;             // NBLOCKS floats (16 KB scratch)

    accnorm_partial_kernel<<<NBLOCKS, NTHREADS, 0, stream>>>(p, pwo, partials);
    reduce_partials_kernel<<<1, NTHREADS, 0, stream>>>(partials, out);
}